// kNNLoss_38620345925716
// MI455X (gfx1250) — compile-verified
//
#include <hip/hip_runtime.h>
#include <hip/hip_bf16.h>
#include <math.h>

// Problem dims (fixed by reference)
#define BATCH   64
#define NPTS    16384
#define NSEED   20
#define KTOP    11        // K+1 smallest distances
#define SEEDPAD 32        // 20 seeds padded to two 16-row WMMA tiles

// Kernel-2 tiling
#define NSPLIT  8                 // segments per batch (occupancy: 512 blocks)
#define SEGPTS  (NPTS / NSPLIT)   // 2048 points per segment
#define CHUNK   256               // points per LDS staging chunk
#define CP      257               // padded row stride (bank-conflict free)
#define NTHREADS 256

typedef __attribute__((ext_vector_type(2))) float v2f;
typedef __attribute__((ext_vector_type(8))) float v8f;

// ---------------------------------------------------------------------------
// Kernel 1: farthest point sampling. One block per batch.
// Seed list semantics match lax.scan: seed[it] = farthest *before* the update,
// so seed[0] == point 0.
// ---------------------------------------------------------------------------
__global__ __launch_bounds__(NTHREADS) void fps_kernel(const float* __restrict__ pcs,
                                                       float* __restrict__ seedxyz) {
    const int b = blockIdx.x;
    const int t = threadIdx.x;
    const float* P = pcs + (size_t)b * NPTS * 3;

    __shared__ float s_val[NTHREADS];
    __shared__ int   s_idx[NTHREADS];
    __shared__ int   s_far;

    // Per-thread slice of the running min-distance array (strided by 256).
    float dist[NPTS / NTHREADS];
#pragma unroll
    for (int k = 0; k < NPTS / NTHREADS; ++k) dist[k] = 1e10f;

    int farthest = 0;
    for (int it = 0; it < NSEED; ++it) {
        // Centroid coords (uniform across block -> scalar loads).
        const float cx = P[farthest * 3 + 0];
        const float cy = P[farthest * 3 + 1];
        const float cz = P[farthest * 3 + 2];
        if (t == 0) {
            float* s = seedxyz + ((size_t)b * NSEED + it) * 3;
            s[0] = cx; s[1] = cy; s[2] = cz;
        }

        float best = -1.0f;
        int   bidx = 0;
#pragma unroll
        for (int k = 0; k < NPTS / NTHREADS; ++k) {
            const int i = t + NTHREADS * k;
            const float dx = P[i * 3 + 0] - cx;
            const float dy = P[i * 3 + 1] - cy;
            const float dz = P[i * 3 + 2] - cz;
            const float d  = dx * dx + dy * dy + dz * dz;
            const float nd = fminf(dist[k], d);
            dist[k] = nd;
            if (nd > best) { best = nd; bidx = i; }   // strict > keeps lowest index
        }

        s_val[t] = best;
        s_idx[t] = bidx;
        __syncthreads();
        // Tree argmax, first-index tie-break (matches jnp.argmax).
        for (int st = NTHREADS / 2; st > 0; st >>= 1) {
            if (t < st) {
                const float v2 = s_val[t + st];
                const int   i2 = s_idx[t + st];
                if (v2 > s_val[t] || (v2 == s_val[t] && i2 < s_idx[t])) {
                    s_val[t] = v2; s_idx[t] = i2;
                }
            }
            __syncthreads();
        }
        if (t == 0) s_far = s_idx[0];
        __syncthreads();
        farthest = s_far;
        __syncthreads();
    }
}

// Branchless ascending insert of d into reg-resident top-KTOP list.
__device__ __forceinline__ void topk_insert(float (&best)[KTOP], float d) {
    if (d < best[KTOP - 1]) {
        float cur = d;
#pragma unroll
        for (int i = 0; i < KTOP; ++i) {
            const float bi = best[i];
            best[i] = fminf(bi, cur);
            cur     = fmaxf(bi, cur);
        }
    }
}

// ---------------------------------------------------------------------------
// Kernel 2: seed-to-point distances via V_WMMA_F32_16X16X4_F32 + partial
// top-11.  One block per (batch, point-segment): grid = BATCH * NSPLIT.
// dist^2(m,n) = |s_m|^2 + |p_n|^2 - 2 * (S . P^T)(m,n)
// A = seeds [16x4] (two M-tiles), B = points [4x16], K = {x,y,z,0}.
// Lane layout (ISA 7.12.2): A/B lane l, vgpr r -> K = 2*(l>>4) + r;
// D lane l, vgpr r -> M = r + 8*(l>>4), N = l&15.
// Partial lists stored as squared distances; sqrt applied at final merge.
// ---------------------------------------------------------------------------
__global__ __launch_bounds__(NTHREADS) void knn_topk_kernel(const float* __restrict__ pcs,
                                                            const float* __restrict__ seedxyz,
                                                            float* __restrict__ part) {
    const int b   = blockIdx.x / NSPLIT;
    const int seg = blockIdx.x % NSPLIT;
    const int t   = threadIdx.x;
    const float* P = pcs + (size_t)b * NPTS * 3;

    __shared__ float s_seed[SEEDPAD][3];
    __shared__ float s_snorm[SEEDPAD];
    __shared__ float s_dist[SEEDPAD * CP];
    __shared__ float s_lists[NTHREADS][KTOP];

    if (t < SEEDPAD) {
        float sx = 0.f, sy = 0.f, sz = 0.f;
        if (t < NSEED) {
            const float* s = seedxyz + ((size_t)b * NSEED + t) * 3;
            sx = s[0]; sy = s[1]; sz = s[2];
        }
        s_seed[t][0] = sx; s_seed[t][1] = sy; s_seed[t][2] = sz;
        s_snorm[t] = sx * sx + sy * sy + sz * sz;
    }
    __syncthreads();

    const int lane = t & 31;
    const int wave = t >> 5;
    const int half = lane >> 4;
    const int l16  = lane & 15;

    // Loop-invariant A-matrix fragments (seed tiles 0 and 1).
    v2f a0, a1;
    a0.x = half ? s_seed[l16][2]      : s_seed[l16][0];
    a0.y = half ? 0.f                 : s_seed[l16][1];
    a1.x = half ? s_seed[16 + l16][2] : s_seed[16 + l16][0];
    a1.y = half ? 0.f                 : s_seed[16 + l16][1];

    // Loop-invariant |seed|^2 per output row this lane owns.
    float sn0[8], sn1[8];
#pragma unroll
    for (int r = 0; r < 8; ++r) {
        sn0[r] = s_snorm[8 * half + r];
        sn1[r] = s_snorm[16 + 8 * half + r];
    }

    // Phase-B assignment: 8 threads per seed row.
    const int s = t & 31;   // seed row in staging buffer
    const int j = t >> 5;   // sub-scanner 0..7
    float best[KTOP];
#pragma unroll
    for (int i = 0; i < KTOP; ++i) best[i] = 3.4e38f;

    const int segBase = seg * SEGPTS;
    for (int base = segBase; base < segBase + SEGPTS; base += CHUNK) {
        // ---- Phase A: each wave fills 2 point-tiles of the chunk via WMMA ----
#pragma unroll
        for (int tt = 0; tt < 2; ++tt) {
            const int tile = wave * 2 + tt;          // 0..15
            const int pi   = base + tile * 16 + l16; // global point index
            const float px = P[pi * 3 + 0];
            const float py = P[pi * 3 + 1];
            const float pz = P[pi * 3 + 2];
            const float pn = px * px + py * py + pz * pz;

            v2f bb;
            bb.x = half ? pz : px;
            bb.y = half ? 0.f : py;

            v8f c0 = {}; v8f c1 = {};
            c0 = __builtin_amdgcn_wmma_f32_16x16x4_f32(false, a0, false, bb,
                                                       (short)0, c0, false, false);
            c1 = __builtin_amdgcn_wmma_f32_16x16x4_f32(false, a1, false, bb,
                                                       (short)0, c1, false, false);

            const int col = tile * 16 + l16;         // column within chunk
#pragma unroll
            for (int r = 0; r < 8; ++r) {
                s_dist[(8 * half + r) * CP + col]      = (sn0[r] + pn) - 2.0f * c0[r];
                s_dist[(16 + 8 * half + r) * CP + col] = (sn1[r] + pn) - 2.0f * c1[r];
            }
        }
        __syncthreads();

        // ---- Phase B: scan staged distances into register top-11 lists ----
        if (s < NSEED) {
#pragma unroll 4
            for (int k = 0; k < CHUNK / 8; ++k) {
                topk_insert(best, s_dist[s * CP + j + 8 * k]);
            }
        }
        __syncthreads();
    }

    // Merge the 8 partial scanner lists per seed; emit segment-partial list.
#pragma unroll
    for (int i = 0; i < KTOP; ++i) s_lists[t][i] = best[i];
    __syncthreads();

    if (t < NSEED) {
        float fin[KTOP];
#pragma unroll
        for (int i = 0; i < KTOP; ++i) fin[i] = 3.4e38f;
        for (int u = 0; u < 8; ++u) {
#pragma unroll
            for (int i = 0; i < KTOP; ++i) topk_insert(fin, s_lists[t + 32 * u][i]);
        }
        float* out = part + (((size_t)b * NSPLIT + seg) * NSEED + t) * KTOP;
#pragma unroll
        for (int i = 0; i < KTOP; ++i) out[i] = fin[i];   // squared distances
    }
}

// ---------------------------------------------------------------------------
// Kernel 2b: merge NSPLIT partial lists per (batch, seed); sqrt + clamp.
// One thread per (batch, seed) row; 1280 rows total.
// ---------------------------------------------------------------------------
__global__ __launch_bounds__(NTHREADS) void knn_merge_kernel(const float* __restrict__ part,
                                                             float* __restrict__ topd) {
    const int row = blockIdx.x * NTHREADS + threadIdx.x;   // b*NSEED + seed
    if (row >= BATCH * NSEED) return;
    const int b  = row / NSEED;
    const int sd = row % NSEED;

    float fin[KTOP];
#pragma unroll
    for (int i = 0; i < KTOP; ++i) fin[i] = 3.4e38f;

    for (int seg = 0; seg < NSPLIT; ++seg) {
        const float* p = part + (((size_t)b * NSPLIT + seg) * NSEED + sd) * KTOP;
#pragma unroll
        for (int i = 0; i < KTOP; ++i) topk_insert(fin, p[i]);
    }
    float* out = topd + (size_t)row * KTOP;
#pragma unroll
    for (int i = 0; i < KTOP; ++i) out[i] = sqrtf(fmaxf(fin[i], 0.0f));
}

// ---------------------------------------------------------------------------
// Kernel 3: final statistics (single block).
//   overall_mean = mean(topd[:, :, 1:]);  m = rowsum / (11 * overall_mean)
//   var = (Sum m^2 - (Sum m)^2 / n) / (n - 1),  n = 1280  (ddof=1)
// ---------------------------------------------------------------------------
__global__ __launch_bounds__(NTHREADS) void knn_var_kernel(const float* __restrict__ topd,
                                                           float* __restrict__ out) {
    const int t = threadIdx.x;
    const int NROWS = BATCH * NSEED;

    __shared__ double sh1[NTHREADS], shA[NTHREADS], shB[NTHREADS];
    double s1 = 0.0, sA = 0.0, sB = 0.0;

    for (int row = t; row < NROWS; row += NTHREADS) {
        const float* r = topd + (size_t)row * KTOP;
        double rs = 0.0, tail = 0.0;
#pragma unroll
        for (int i = 0; i < KTOP; ++i) {
            const double v = (double)r[i];
            rs += v;
            if (i > 0) tail += v;
        }
        s1 += tail;
        sA += rs;
        sB += rs * rs;
    }
    sh1[t] = s1; shA[t] = sA; shB[t] = sB;
    __syncthreads();
    for (int st = NTHREADS / 2; st > 0; st >>= 1) {
        if (t < st) {
            sh1[t] += sh1[t + st];
            shA[t] += shA[t + st];
            shB[t] += shB[t + st];
        }
        __syncthreads();
    }
    if (t == 0) {
        const double n    = (double)NROWS;
        const double mean = sh1[0] / (n * (double)(KTOP - 1));   // mean over [:, :, 1:]
        const double c    = 1.0 / ((double)KTOP * mean);
        const double sm   = c * shA[0];          // Sum m
        const double sm2  = c * c * shB[0];      // Sum m^2
        const double var  = (sm2 - sm * sm / n) / (n - 1.0);
        out[0] = (float)var;
    }
}

// ---------------------------------------------------------------------------
extern "C" void kernel_launch(void* const* d_in, const int* in_sizes, int n_in,
                              void* d_out, int out_size, void* d_ws, size_t ws_size,
                              hipStream_t stream) {
    (void)in_sizes; (void)n_in; (void)out_size; (void)ws_size;
    const float* pcs = (const float*)d_in[0];
    float* outp = (float*)d_out;

    // Workspace layout (bytes):
    //   [0,      15360)  : seed coords, float[64][20][3]
    //   [16384, 466944)  : partial top lists, float[64][8][20][11] (squared)
    //   [475136, 531456) : final top distances, float[1280][11]
    float* seedxyz = (float*)d_ws;
    float* part    = (float*)((char*)d_ws + 16384);
    float* topd    = (float*)((char*)d_ws + 475136);

    fps_kernel<<<BATCH, NTHREADS, 0, stream>>>(pcs, seedxyz);
    knn_topk_kernel<<<BATCH * NSPLIT, NTHREADS, 0, stream>>>(pcs, seedxyz, part);
    knn_merge_kernel<<<(BATCH * NSEED + NTHREADS - 1) / NTHREADS, NTHREADS, 0, stream>>>(part, topd);
    knn_var_kernel<<<1, NTHREADS, 0, stream>>>(topd, outp);
}